// StandardGCN_40046275068307
// MI455X (gfx1250) — compile-verified
//
#include <hip/hip_runtime.h>

// Problem constants (match reference)
#define N_NODES 100000
#define EDGES   1600000
#define D_IN    128
#define D_HID   128
#define D_OUT   64
#define BN_EPS  1e-5f

typedef float v2f __attribute__((ext_vector_type(2)));
typedef float v8f __attribute__((ext_vector_type(8)));
typedef unsigned int u32x4 __attribute__((ext_vector_type(4)));
typedef int i32x4 __attribute__((ext_vector_type(4)));
typedef int i32x8 __attribute__((ext_vector_type(8)));

// ---------------------------------------------------------------------------
// zero-fill (float4 granularity, grid-stride)
// ---------------------------------------------------------------------------
__global__ void zero_kernel(float4* __restrict__ p, int n4) {
  int i = blockIdx.x * blockDim.x + threadIdx.x;
  int stride = gridDim.x * blockDim.x;
  float4 z = {0.f, 0.f, 0.f, 0.f};
  for (; i < n4; i += stride) p[i] = z;
}

// ---------------------------------------------------------------------------
// GEMM: H[N,F] = act(A[N,128]) @ W[128,F]
// One wave computes one 16x16 output tile with V_WMMA_F32_16X16X4_F32.
// The 16x128 f32 A tile (8 KB) is DMA'd into LDS once per block by the
// Tensor Data Mover (tensor_load_to_lds, TENSORcnt), removing the 8x
// redundant per-wave global loads of A. W (<=64 KB) stays on the direct
// global path (L1/L2 resident across all 6250 blocks).
//
// f32 16x16x4 fragment layouts (ISA 7.12.2):
//  A (16x4, MxK): lane L holds M=L%16, K = 2*(L/16) + {0,1} in its 2 regs
//  B (4x16, KxN): lane L holds N=L%16, K = 2*(L/16) + {0,1} in its 2 regs
//  C/D (16x16):   reg r, lane L -> M = r + 8*(L/16), N = L%16
// ---------------------------------------------------------------------------
template <int F, bool BN>
__global__ __launch_bounds__((F / 16) * 32)
void gemm_wmma_kernel(const float* __restrict__ A, const float* __restrict__ W,
                      const float* __restrict__ scale, const float* __restrict__ shift,
                      float* __restrict__ H) {
  extern __shared__ float As[];            // [16][128], dynamic => LDS offset 0
  const int lane = threadIdx.x & 31;
  const int wave = threadIdx.x >> 5;       // column tile within row tile
  const int m0   = blockIdx.x * 16;
  const int n0   = wave * 16;
  const int mr   = lane & 15;              // M (for A) or N (for B/C)
  const int kg   = (lane >> 4) * 2;        // K sub-offset: 0 or 2

  if (threadIdx.x < 32) {
    // Tensor DMA descriptor: 2-D tile, 16 rows x 128 cols of f32,
    // row stride 128 elements, into LDS offset 0.
    unsigned long long ga = (unsigned long long)(const void*)(A + (size_t)m0 * 128);
    u32x4 g0;
    g0[0] = 1u;                                   // count=1, user descriptor
    g0[1] = 0u;                                   // lds_addr = 0 (dynamic LDS base)
    g0[2] = (unsigned int)ga;                     // global_addr[31:0]
    g0[3] = ((unsigned int)(ga >> 32) & 0x01FFFFFFu) | (2u << 30);  // addr[56:32] | type=2
    i32x8 g1;
    g1[0] = 2 << 16;                              // data_size = 4 bytes
    g1[1] = 128 << 16;                            // tensor_dim0 = 128 (low 16 in [31:16])
    g1[2] = 16 << 16;                             // tensor_dim0 hi=0 | tensor_dim1 = 16
    g1[3] = 128 << 16;                            // tensor_dim1 hi=0 | tile_dim0 = 128
    g1[4] = 16;                                   // tile_dim1 = 16, tile_dim2 = 0
    g1[5] = 128;                                  // tensor_dim0_stride = 128 elements
    g1[6] = 0;
    g1[7] = 0;
    i32x4 z4 = {0, 0, 0, 0};                      // groups 2/3 unused (2-D tensor)
    i32x8 z8 = {0, 0, 0, 0, 0, 0, 0, 0};          // 6-arg toolchain extra group
    __builtin_amdgcn_tensor_load_to_lds(g0, g1, z4, z4, z8, 0);
    __builtin_amdgcn_s_wait_tensorcnt(0);
  }
  __syncthreads();                                // publish LDS tile to all waves

  v8f acc = {0.f, 0.f, 0.f, 0.f, 0.f, 0.f, 0.f, 0.f};
  const float* __restrict__ arow = As + mr * 128;

#pragma unroll 4
  for (int k = 0; k < 128; k += 4) {
    float x0 = arow[k + kg];                      // ds_load from the TDM-filled tile
    float x1 = arow[k + kg + 1];
    if (BN) {
      x0 = fmaxf(fmaf(x0, scale[k + kg],     shift[k + kg]),     0.f);
      x1 = fmaxf(fmaf(x1, scale[k + kg + 1], shift[k + kg + 1]), 0.f);
    }
    v2f a, b;
    a.x = x0;
    a.y = x1;
    b.x = W[(size_t)(k + kg)     * F + n0 + mr];
    b.y = W[(size_t)(k + kg + 1) * F + n0 + mr];
    acc = __builtin_amdgcn_wmma_f32_16x16x4_f32(
        /*neg_a=*/false, a, /*neg_b=*/false, b,
        /*c_mod=*/(short)0, acc, /*reuse_a=*/false, /*reuse_b=*/false);
  }

  const int n     = n0 + mr;
  const int mbase = m0 + (lane >> 4) * 8;
#pragma unroll
  for (int r = 0; r < 8; ++r)
    H[(size_t)(mbase + r) * F + n] = acc[r];
}

// ---------------------------------------------------------------------------
// Edge scatter: AGG[dst[e], :] += w[e] * H[src[e], :]
// FSHIFT = log2(F/4). For F=128 all 32 lanes of a wave work the same edge ->
// scalar-friendly metadata loads, coalesced b128 gather, coalesced atomics.
// AGG lives in L2 (51.2 MB < 192 MB), so the f32 atomics are L2-resident.
// ---------------------------------------------------------------------------
template <int FSHIFT>
__global__ __launch_bounds__(256)
void scatter_kernel(const float* __restrict__ H, const int* __restrict__ src,
                    const int* __restrict__ dst, const float* __restrict__ ew,
                    float* __restrict__ agg) {
  const int F = 4 << FSHIFT;
  int tid = blockIdx.x * blockDim.x + threadIdx.x;
  if (tid >= (EDGES << FSHIFT)) return;
  const int e = tid >> FSHIFT;
  const int c = (tid & ((1 << FSHIFT) - 1)) << 2;   // feature group offset
  const float w = ew[e];
  const float4 h = *(const float4*)(H + (size_t)src[e] * F + c);
  float* op = agg + (size_t)dst[e] * F + c;
  atomicAdd(op + 0, w * h.x);
  atomicAdd(op + 1, w * h.y);
  atomicAdd(op + 2, w * h.z);
  atomicAdd(op + 3, w * h.w);
}

// ---------------------------------------------------------------------------
// Per-column sum and sum-of-squares over the node dimension (for BN stats).
// blockDim.x == F; thread c owns column c, rows strided by gridDim.
// ---------------------------------------------------------------------------
__global__ void stats_kernel(const float* __restrict__ X, float* __restrict__ sum,
                             float* __restrict__ sumsq, int F) {
  const int c = threadIdx.x;
  float s = 0.f, s2 = 0.f;
  for (int r = blockIdx.x; r < N_NODES; r += gridDim.x) {
    float x = X[(size_t)r * F + c];
    s += x;
    s2 += x * x;
  }
  atomicAdd(&sum[c], s);
  atomicAdd(&sumsq[c], s2);
}

// ---------------------------------------------------------------------------
// BN params: scale = gamma * rsqrt(var+eps), shift = beta - mean*scale.
// (conv bias b cancels exactly: BN subtracts the column mean.)
// ---------------------------------------------------------------------------
__global__ void bnparam_kernel(const float* __restrict__ sum, const float* __restrict__ sumsq,
                               const float* __restrict__ gamma, const float* __restrict__ beta,
                               float* __restrict__ scale, float* __restrict__ shift) {
  const int c = threadIdx.x;
  const float invN = 1.0f / (float)N_NODES;
  float mu  = sum[c] * invN;
  float var = sumsq[c] * invN - mu * mu;
  float s   = gamma[c] * rsqrtf(var + BN_EPS);
  scale[c] = s;
  shift[c] = beta[c] - mu * s;
}

// d_out[f] = N * b2[f]   (the surviving part of the final conv bias)
__global__ void initout_kernel(float* __restrict__ out, const float* __restrict__ b2) {
  out[threadIdx.x] = (float)N_NODES * b2[threadIdx.x];
}

// d_out[f] += sum over nodes of AGG2[n,f]
__global__ void colsum_kernel(const float* __restrict__ X, float* __restrict__ out) {
  const int c = threadIdx.x;   // blockDim = D_OUT
  float s = 0.f;
  for (int r = blockIdx.x; r < N_NODES; r += gridDim.x)
    s += X[(size_t)r * D_OUT + c];
  atomicAdd(&out[c], s);
}

// ---------------------------------------------------------------------------
// Orchestration
// ---------------------------------------------------------------------------
extern "C" void kernel_launch(void* const* d_in, const int* in_sizes, int n_in,
                              void* d_out, int out_size, void* d_ws, size_t ws_size,
                              hipStream_t stream) {
  (void)in_sizes; (void)n_in; (void)out_size; (void)ws_size;

  const float* nf  = (const float*)d_in[0];
  const int*   ei  = (const int*)d_in[1];     // [2, E]: row0 = src, row1 = dst
  const float* ew  = (const float*)d_in[2];
  const float* W0  = (const float*)d_in[3];
  // b0 = d_in[4]  (cancels inside BN0)
  const float* W1  = (const float*)d_in[5];
  // b1 = d_in[6]  (cancels inside BN1)
  const float* W2  = (const float*)d_in[7];
  const float* b2  = (const float*)d_in[8];
  const float* g0  = (const float*)d_in[9];
  const float* be0 = (const float*)d_in[10];
  const float* g1  = (const float*)d_in[11];
  const float* be1 = (const float*)d_in[12];
  float* out = (float*)d_out;

  const int* src = ei;
  const int* dst = ei + EDGES;

  // Workspace layout (floats): bufA[N*128] | bufB[N*128] | sum[128] | sumsq[128] | scale[128] | shift[128]
  float* bufA  = (float*)d_ws;
  float* bufB  = bufA + (size_t)N_NODES * 128;
  float* sum   = bufB + (size_t)N_NODES * 128;
  float* sumsq = sum + 128;
  float* scale = sum + 256;
  float* shift = sum + 384;

  const int rowTiles   = N_NODES / 16;              // 6250
  const int zeroGrid   = 2048;
  const int statGrid   = 512;
  const int scat128Blk = (EDGES * 32) / 256;        // F=128: E*32 threads
  const int scat64Blk  = (EDGES * 16) / 256;        // F=64:  E*16 threads
  const int nElems128  = N_NODES * 128;
  const int ldsBytes   = 16 * 128 * 4;              // 8 KB A tile

  // ----- layer 0: H = X @ W0 ; AGG = scatter ; BN stats -----
  gemm_wmma_kernel<128, false><<<rowTiles, 256, ldsBytes, stream>>>(nf, W0, nullptr, nullptr, bufA);
  zero_kernel<<<zeroGrid, 256, 0, stream>>>((float4*)bufB, nElems128 / 4);
  zero_kernel<<<1, 64, 0, stream>>>((float4*)sum, 64);   // sum+sumsq (256 floats)
  scatter_kernel<5><<<scat128Blk, 256, 0, stream>>>(bufA, src, dst, ew, bufB);
  stats_kernel<<<statGrid, 128, 0, stream>>>(bufB, sum, sumsq, 128);
  bnparam_kernel<<<1, 128, 0, stream>>>(sum, sumsq, g0, be0, scale, shift);

  // ----- layer 1: H = relu(BN(AGG)) @ W1 ; AGG = scatter ; BN stats -----
  gemm_wmma_kernel<128, true><<<rowTiles, 256, ldsBytes, stream>>>(bufB, W1, scale, shift, bufA);
  zero_kernel<<<zeroGrid, 256, 0, stream>>>((float4*)bufB, nElems128 / 4);
  zero_kernel<<<1, 64, 0, stream>>>((float4*)sum, 64);
  scatter_kernel<5><<<scat128Blk, 256, 0, stream>>>(bufA, src, dst, ew, bufB);
  stats_kernel<<<statGrid, 128, 0, stream>>>(bufB, sum, sumsq, 128);
  bnparam_kernel<<<1, 128, 0, stream>>>(sum, sumsq, g1, be1, scale, shift);

  // ----- layer 2: H = relu(BN(AGG)) @ W2 ; AGG2 = scatter ; out = colsum + N*b2 -----
  gemm_wmma_kernel<64, true><<<rowTiles, 128, ldsBytes, stream>>>(bufB, W2, scale, shift, bufA);
  zero_kernel<<<zeroGrid, 256, 0, stream>>>((float4*)bufB, (N_NODES * 64) / 4);
  initout_kernel<<<1, 64, 0, stream>>>(out, b2);
  scatter_kernel<4><<<scat64Blk, 256, 0, stream>>>(bufA, src, dst, ew, bufB);
  colsum_kernel<<<statGrid, 64, 0, stream>>>(bufB, out);
}